// OD2Score_48782238548108
// MI455X (gfx1250) — compile-verified
//
#include <hip/hip_runtime.h>

#define N_PRED   2048
#define BLK      1024
#define RAW_F    (N_PRED * 6)            // 12288 floats = 48 KB per batch
#define CONF_TH  0.01f
#define IOU_TH   0.5f
#define CLS_OFF  4096.0f

// LDS layout (phased aliasing inside one 64 KB block)
#define SK_OFF   49152                   // u64[2048]  16 KB   (phase A+B)
#define SBOX_OFF 0                       // float4[2048] 32 KB (phase B, overlaps raw)
#define SUP_OFF  32768                   // u8[2048]    2 KB   (phase B)
#define RED_OFF  34816                   // float[32]          (phase B)

#define AS1 __attribute__((address_space(1)))
#define AS3 __attribute__((address_space(3)))

typedef __attribute__((ext_vector_type(4))) int v4i;

#if defined(__has_builtin)
#if __has_builtin(__builtin_amdgcn_global_load_async_to_lds_b128)
#define HAVE_ASYNC_LDS 1
#endif
#endif

__device__ __forceinline__ float iou_pair(float ax1, float ay1, float ax2, float ay2, float aarea,
                                          float bx1, float by1, float bx2, float by2, float barea) {
    float iw = fmaxf(fminf(ax2, bx2) - fmaxf(ax1, bx1), 0.0f);
    float ih = fmaxf(fminf(ay2, by2) - fmaxf(ay1, by1), 0.0f);
    float inter = iw * ih;
    return inter / (aarea + barea - inter);   // denom > 0 always (areas >= 25)
}

__global__ __launch_bounds__(BLK) void od2score_kernel(const float* __restrict__ pred,
                                                       const float* __restrict__ tgt,
                                                       float* __restrict__ out) {
    __shared__ __align__(16) unsigned char smem[65536];
    float*              raw  = (float*)smem;                          // phase A: raw batch
    unsigned long long* sk   = (unsigned long long*)(smem + SK_OFF);  // sort keys
    float4*             sbox = (float4*)(smem + SBOX_OFF);            // phase B: shifted boxes
    unsigned char*      sup  = smem + SUP_OFF;                        // phase B: 0=cand 1=supp 2=invalid
    float*              red  = (float*)(smem + RED_OFF);              // phase B: reduction

    const int b = blockIdx.x;
    const int t = threadIdx.x;
    const float* __restrict__ p = pred + (size_t)b * RAW_F;

    // ---- Phase 0: stage the whole batch (48 KB) into LDS
#if defined(HAVE_ASYNC_LDS)
    // CDNA5 async-to-LDS path: 3 x b128 per lane, tracked by ASYNCcnt (no VGPR round-trip)
    #pragma unroll
    for (int k = 0; k < 3; ++k) {
        int byteOff = (t + k * BLK) * 16;
        __builtin_amdgcn_global_load_async_to_lds_b128(
            (AS1 v4i*)(unsigned long long)((const char*)p + byteOff),
            (AS3 v4i*)(unsigned int)(unsigned long long)((const char*)raw + byteOff),
            0, 0);
    }
#if __has_builtin(__builtin_amdgcn_s_wait_asynccnt)
    __builtin_amdgcn_s_wait_asynccnt(0);
#else
    asm volatile("s_wait_asynccnt 0x0" ::: "memory");
#endif
#else
    if (t < (RAW_F * 4 + 255) / 256) {
        __builtin_prefetch((const char*)p + t * 256, 0, 1);
    }
    for (int i = t; i < RAW_F; i += BLK) raw[i] = p[i];
#endif
    __syncthreads();

    // target box (uniform)
    const float tx1 = tgt[0], ty1 = tgt[1], tx2 = tgt[2], ty2 = tgt[3];
    const float tarea = (tx2 - tx1) * (ty2 - ty1);

    // ---- Phase 1: build sortable keys: stable descending conf, invalid -> -inf (last)
    for (int i = t; i < N_PRED; i += BLK) {
        float conf = raw[i * 6 + 4];
        bool valid = conf > CONF_TH;
        unsigned int bits = valid ? __float_as_uint(conf) : 0xFF800000u;  // -inf
        unsigned int mapped = (bits & 0x80000000u) ? ~bits : (bits | 0x80000000u);
        unsigned int dk = ~mapped;  // ascending dk == descending key
        sk[i] = ((unsigned long long)dk << 32) | (unsigned int)i;
    }
    __syncthreads();

    // ---- Phase 2: bitonic sort (ascending u64) fully in LDS
    for (int k = 2; k <= N_PRED; k <<= 1) {
        for (int j = k >> 1; j > 0; j >>= 1) {
            for (int e = t; e < N_PRED; e += BLK) {
                int ixj = e ^ j;
                if (ixj > e) {
                    bool up = ((e & k) == 0);
                    unsigned long long a = sk[e], c = sk[ixj];
                    if ((a > c) == up) { sk[e] = c; sk[ixj] = a; }
                }
            }
            __syncthreads();
        }
    }

    // ---- Phase 3a: gather both sorted slots from LDS-raw into registers
    float gx1[2], gy1[2], gx2[2], gy2[2], gcls[2], gkey[2];
    #pragma unroll
    for (int q = 0; q < 2; ++q) {
        int i = t + q * BLK;
        unsigned long long s = sk[i];
        int oi = (int)(unsigned int)(s & 0xFFFFFFFFu);
        unsigned int mapped = ~(unsigned int)(s >> 32);
        unsigned int kb = (mapped & 0x80000000u) ? (mapped & 0x7FFFFFFFu) : ~mapped;
        gkey[q] = __uint_as_float(kb);        // == conf if valid, -inf if invalid
        gx1[q] = raw[oi * 6 + 0];
        gy1[q] = raw[oi * 6 + 1];
        gx2[q] = raw[oi * 6 + 2];
        gy2[q] = raw[oi * 6 + 3];
        gcls[q] = raw[oi * 6 + 5];
    }
    __syncthreads();   // all raw reads complete before the region is repurposed

    // ---- Phase 3b: write shifted boxes + state; keep per-slot score in registers
    float4 bb[2];
    float  myarea[2], mysc[2];
    int    st[2];
    #pragma unroll
    for (int q = 0; q < 2; ++q) {
        int i = t + q * BLK;
        bool valid = gkey[q] > CONF_TH;
        float off = gcls[q] * CLS_OFF;
        float4 v = make_float4(gx1[q] + off, gy1[q] + off, gx2[q] + off, gy2[q] + off);
        bb[q] = v;
        myarea[q] = (v.z - v.x) * (v.w - v.y);
        // IoU with (unshifted) target box
        float iw = fmaxf(fminf(gx2[q], tx2) - fmaxf(gx1[q], tx1), 0.0f);
        float ih = fmaxf(fminf(gy2[q], ty2) - fmaxf(gy1[q], ty1), 0.0f);
        float inter = iw * ih;
        float area = (gx2[q] - gx1[q]) * (gy2[q] - gy1[q]);
        float tiou = inter / (tarea + area - inter);
        mysc[q] = valid ? tiou * gkey[q] : 0.0f;
        st[q]   = valid ? 0 : 2;
        sbox[i] = v;
        sup[i]  = (unsigned char)st[q];
    }
    __syncthreads();

    // ---- Phase 4: greedy class-aware NMS (sequential over sorted order)
    const int j1 = t + BLK;
    for (int i = 0; i < N_PRED; ++i) {
        int s = sup[i];                   // uniform LDS broadcast
        if (s == 2) break;                // sorted: everything after is invalid
        if (s == 0) {                     // box i is kept -> suppress overlapping later boxes
            float4 bi = sbox[i];
            float ai = (bi.z - bi.x) * (bi.w - bi.y);
            if (st[0] == 0 && t > i) {
                if (iou_pair(bi.x, bi.y, bi.z, bi.w, ai,
                             bb[0].x, bb[0].y, bb[0].z, bb[0].w, myarea[0]) > IOU_TH) {
                    st[0] = 1; sup[t] = 1;
                }
            }
            if (st[1] == 0 && j1 > i) {
                if (iou_pair(bi.x, bi.y, bi.z, bi.w, ai,
                             bb[1].x, bb[1].y, bb[1].z, bb[1].w, myarea[1]) > IOU_TH) {
                    st[1] = 1; sup[j1] = 1;
                }
            }
            __syncthreads();              // uniform branch: all waves participate
        }
    }

    // ---- Phase 5: max over kept of tiou*conf (wave32 shuffle + LDS reduce)
    float m = 0.0f;
    if (st[0] == 0) m = fmaxf(m, mysc[0]);
    if (st[1] == 0) m = fmaxf(m, mysc[1]);
    #pragma unroll
    for (int o = 16; o > 0; o >>= 1) m = fmaxf(m, __shfl_xor(m, o, 32));
    if ((t & 31) == 0) red[t >> 5] = m;
    __syncthreads();
    if (t < 32) {
        float v = red[t];
        #pragma unroll
        for (int o = 16; o > 0; o >>= 1) v = fmaxf(v, __shfl_xor(v, o, 32));
        if (t == 0) out[b] = v;
    }
}

extern "C" void kernel_launch(void* const* d_in, const int* in_sizes, int n_in,
                              void* d_out, int out_size, void* d_ws, size_t ws_size,
                              hipStream_t stream) {
    const float* pred = (const float*)d_in[0];   // [B, 2048, 6] float32
    const float* tgt  = (const float*)d_in[1];   // [1, 4]      float32
    float* out = (float*)d_out;                  // [B]         float32
    int B = in_sizes[0] / RAW_F;
    od2score_kernel<<<dim3(B), dim3(BLK), 0, stream>>>(pred, tgt, out);
}